// ScaledMixtralSparseMoeBlock_20280835571725
// MI455X (gfx1250) — compile-verified
//
#include <hip/hip_runtime.h>
#include <math.h>

#define HIDDEN 2048
#define FFN    7168
#define NEXP   8
#define NTOK   8192            // B*S
#define MTILE  32              // tokens per block (2 WMMA M-subtiles)
#define NWAVE  16              // 512 threads
#define CHUNK  256             // FFN columns per block iteration (16 waves * 16)
#define HM_STRIDE 260          // 256 + 4 pad -> stride%64==4, conflict-free frag loads

typedef __attribute__((ext_vector_type(2))) float v2f;
typedef __attribute__((ext_vector_type(8))) float v8f;

#define WMMA_F32(a, b, c) \
    __builtin_amdgcn_wmma_f32_16x16x4_f32(false, (a), false, (b), (short)0, (c), false, false)

// ---------------- kernel 0: zero output + expert counters ----------------
__global__ void zero_kernel(float* __restrict__ out, int n, int* __restrict__ cnt) {
    for (int i = blockIdx.x * blockDim.x + threadIdx.x; i < n; i += gridDim.x * blockDim.x)
        out[i] = 0.0f;
    if (blockIdx.x == 0 && threadIdx.x < NEXP) cnt[threadIdx.x] = 0;
}

// ---------------- kernel 1: router (logits, softmax, top-2, scatter) ----------------
__global__ __launch_bounds__(256)
void router_kernel(const float* __restrict__ x, const float* __restrict__ gw,
                   float* __restrict__ logits_out,
                   int* __restrict__ cnt, int* __restrict__ tok, float* __restrict__ wgt) {
    const int t   = blockIdx.x;
    const int tid = threadIdx.x;
    const float* xr = x + (size_t)t * HIDDEN;

    float acc[NEXP];
#pragma unroll
    for (int e = 0; e < NEXP; ++e) acc[e] = 0.0f;

    for (int h = tid; h < HIDDEN; h += 256) {
        const float xv = xr[h];
#pragma unroll
        for (int e = 0; e < NEXP; ++e) acc[e] += xv * gw[e * HIDDEN + h];
    }

    __shared__ float red[256];
    __shared__ float lg[NEXP];
    for (int e = 0; e < NEXP; ++e) {
        red[tid] = acc[e];
        __syncthreads();
        for (int s = 128; s > 0; s >>= 1) {
            if (tid < s) red[tid] += red[tid + s];
            __syncthreads();
        }
        if (tid == 0) lg[e] = red[0];
        __syncthreads();
    }

    if (tid == 0) {
        float mx = lg[0];
#pragma unroll
        for (int e = 1; e < NEXP; ++e) mx = fmaxf(mx, lg[e]);
        float p[NEXP], sum = 0.0f;
#pragma unroll
        for (int e = 0; e < NEXP; ++e) { p[e] = __expf(lg[e] - mx); sum += p[e]; }
        const float inv = 1.0f / sum;
#pragma unroll
        for (int e = 0; e < NEXP; ++e) p[e] *= inv;

        // top-2, lowest index wins ties (matches jax.lax.top_k)
        int i1 = 0;
#pragma unroll
        for (int e = 1; e < NEXP; ++e) if (p[e] > p[i1]) i1 = e;
        int i2 = (i1 == 0) ? 1 : 0;
#pragma unroll
        for (int e = 0; e < NEXP; ++e) if (e != i1 && p[e] > p[i2]) i2 = e;

        const float s2 = p[i1] + p[i2];
#pragma unroll
        for (int e = 0; e < NEXP; ++e) logits_out[(size_t)t * NEXP + e] = lg[e];

        int pos = atomicAdd(&cnt[i1], 1);
        tok[i1 * NTOK + pos] = t;  wgt[i1 * NTOK + pos] = p[i1] / s2;
        pos = atomicAdd(&cnt[i2], 1);
        tok[i2 * NTOK + pos] = t;  wgt[i2 * NTOK + pos] = p[i2] / s2;
    }
}

// ---------------- kernel 2: fused expert FFN via fp32 WMMA ----------------
__global__ __launch_bounds__(512, 1)
void moe_kernel(const float* __restrict__ x,
                const float* __restrict__ w1, const float* __restrict__ w3,
                const float* __restrict__ w2, const float* __restrict__ scales,
                const int* __restrict__ cnt, const int* __restrict__ tok,
                const float* __restrict__ wgt,
                float* __restrict__ out) {
    const int e    = blockIdx.y;
    const int n    = cnt[e];
    const int base = blockIdx.x * MTILE;
    if (base >= n) return;                    // block-uniform: EXEC stays full below

    extern __shared__ float smem[];
    float* inv_sc = smem;                     // [HIDDEN]   reciprocal scales (broadcast)
    float* hm     = smem + HIDDEN;            // [MTILE][HM_STRIDE] hmid chunk
    __shared__ int   s_tok[MTILE];
    __shared__ float s_w[MTILE];

    const int tid  = threadIdx.x;
    const int lane = tid & 31;
    const int wv   = tid >> 5;                // wave id 0..15

    if (tid < MTILE) {
        const int idx = base + tid;
        s_tok[tid] = (idx < n) ? tok[e * NTOK + idx] : 0;
        s_w[tid]   = (idx < n) ? wgt[e * NTOK + idx] : 0.0f;
    }
    const float* sc = scales + e * HIDDEN;
    for (int i = tid; i < HIDDEN; i += 512) inv_sc[i] = 1.0f / sc[i];
    __syncthreads();

    const float* w1e = w1 + (size_t)e * FFN * HIDDEN;
    const float* w3e = w3 + (size_t)e * FFN * HIDDEN;
    const float* w2e = w2 + (size_t)e * HIDDEN * FFN;

    // 16x4 f32 fragment addressing: lanes 0-15 -> K = k0,k0+1 ; lanes 16-31 -> K = k0+2,k0+3
    const int m16   = lane & 15;
    const int khalf = (lane >> 4) << 1;
    const int mrow  = (lane >> 4) << 3;       // C/D layout row base
    const int ncol  = lane & 15;              // C/D layout column
    const int h0w   = wv * 128;               // this wave's slice of H

    // per-lane x row pointers for the two M-subtiles (xs = x * inv_scale, on the fly)
    const float* xp0 = x + (size_t)s_tok[m16]      * HIDDEN + khalf;
    const float* xp1 = x + (size_t)s_tok[16 + m16] * HIDDEN + khalf;
    const float* ivp = inv_sc + khalf;

    v8f yacc[16] = {};                        // [sub(2)][ht(8)] 16x16 f32 tiles = 128 VGPRs

    for (int chunk = 0; chunk < FFN; chunk += CHUNK) {
        // -------- up-projection: this wave owns 16 FFN columns of the chunk --------
        const int f0 = chunk + wv * 16;
        const float* w1r = w1e + (size_t)(f0 + m16) * HIDDEN + khalf;
        const float* w3r = w3e + (size_t)(f0 + m16) * HIDDEN + khalf;
        v8f aG0 = {}, aU0 = {}, aG1 = {}, aU1 = {};
#pragma unroll 4
        for (int k0 = 0; k0 < HIDDEN; k0 += 4) {
            const v2f s  = *(const v2f*)(ivp + k0);     // LDS broadcast
            const v2f x0 = *(const v2f*)(xp0 + k0) * s; // A frag, subtile 0
            const v2f x1 = *(const v2f*)(xp1 + k0) * s; // A frag, subtile 1
            const v2f b1 = *(const v2f*)(w1r + k0);
            const v2f b3 = *(const v2f*)(w3r + k0);
            aG0 = WMMA_F32(x0, b1, aG0);
            aU0 = WMMA_F32(x0, b3, aU0);
            aG1 = WMMA_F32(x1, b1, aG1);
            aU1 = WMMA_F32(x1, b3, aU1);
        }
        // silu(G)*U
        float hv[16];
#pragma unroll
        for (int r = 0; r < 8; ++r) {
            const float g0 = aG0[r], g1 = aG1[r];
            hv[r]     = (g0 / (1.0f + __expf(-g0))) * aU0[r];
            hv[8 + r] = (g1 / (1.0f + __expf(-g1))) * aU1[r];
        }

        __syncthreads();                      // previous chunk's down-phase reads done
#pragma unroll
        for (int r = 0; r < 8; ++r) {
            hm[(mrow + r)      * HM_STRIDE + wv * 16 + ncol] = hv[r];
            hm[(mrow + r + 16) * HM_STRIDE + wv * 16 + ncol] = hv[8 + r];
        }
        __syncthreads();

        // -------- down-projection: y[32 x 128] += hmid[32 x 256] @ w2^T --------
        // ht-outer / k-inner over one full 128B line segment: every w2 line read once.
        const float* hr0 = hm + m16 * HM_STRIDE + khalf;
        const float* hr1 = hr0 + 16 * HM_STRIDE;
        for (int kseg = 0; kseg < CHUNK; kseg += 32) {
            v2f a0[8], a1[8];                 // cache A frags for this k-segment
#pragma unroll
            for (int j = 0; j < 8; ++j) {
                a0[j] = *(const v2f*)(hr0 + kseg + 4 * j);
                a1[j] = *(const v2f*)(hr1 + kseg + 4 * j);
            }
#pragma unroll
            for (int ht = 0; ht < 8; ++ht) {
                const float* w2r = w2e + (size_t)(h0w + ht * 16 + m16) * FFN
                                 + chunk + kseg + khalf;
#pragma unroll
                for (int j = 0; j < 8; ++j) {
                    const v2f b = *(const v2f*)(w2r + 4 * j);
                    yacc[ht]     = WMMA_F32(a0[j], b, yacc[ht]);
                    yacc[8 + ht] = WMMA_F32(a1[j], b, yacc[8 + ht]);
                }
            }
        }
    }

    // weighted accumulate into out (each token gets exactly 2 expert contributions)
#pragma unroll
    for (int sub = 0; sub < 2; ++sub) {
#pragma unroll
        for (int ht = 0; ht < 8; ++ht) {
            const int h = h0w + ht * 16 + ncol;
#pragma unroll
            for (int r = 0; r < 8; ++r) {
                const int m = sub * 16 + mrow + r;
                atomicAdd(&out[(size_t)s_tok[m] * HIDDEN + h],
                          yacc[sub * 8 + ht][r] * s_w[m]);
            }
        }
    }
}

// ---------------- launch ----------------
extern "C" void kernel_launch(void* const* d_in, const int* in_sizes, int n_in,
                              void* d_out, int out_size, void* d_ws, size_t ws_size,
                              hipStream_t stream) {
    (void)in_sizes; (void)n_in; (void)out_size; (void)ws_size;
    const float* x  = (const float*)d_in[0];
    const float* gw = (const float*)d_in[1];
    const float* w1 = (const float*)d_in[2];
    const float* w3 = (const float*)d_in[3];
    const float* w2 = (const float*)d_in[4];
    const float* sc = (const float*)d_in[5];

    float* out    = (float*)d_out;
    float* logits = out + (size_t)NTOK * HIDDEN;   // outputs concatenated: out, router_logits

    int*   cnt = (int*)d_ws;                        // [8] (padded to 64 ints)
    int*   tok = cnt + 64;                          // [8][8192]
    float* wgt = (float*)(tok + NEXP * NTOK);       // [8][8192]

    zero_kernel<<<4096, 256, 0, stream>>>(out, NTOK * HIDDEN, cnt);
    router_kernel<<<NTOK, 256, 0, stream>>>(x, gw, logits, cnt, tok, wgt);

    const size_t lds = (size_t)(HIDDEN + MTILE * HM_STRIDE) * sizeof(float); // ~41 KB
    dim3 grid(NTOK / MTILE, NEXP);                  // worst case: all tokens on one expert
    moe_kernel<<<grid, 512, lds, stream>>>(x, w1, w3, w2, sc, cnt, tok, wgt, out);
}